// SSIM_9320079032334
// MI455X (gfx1250) — compile-verified
//
#include <hip/hip_runtime.h>

typedef float v2f __attribute__((ext_vector_type(2)));
typedef float v8f __attribute__((ext_vector_type(8)));

#define W 512
#define H 512
#define OW 505
#define OH 505
#define ROWSTRIDE 516                 // padded LDS row stride in floats (bank-conflict free)
#define QSTRIDE (16 * ROWSTRIDE)      // per-quantity LDS chunk (16 rows)
#define SSIM_C1 1.0e-4f
#define SSIM_C2 9.0e-4f
#define INV64 (1.0f / 64.0f)
#define LDS_FLOATS (5 * QSTRIDE + 16) // + pad so halo b64 reads stay in-bounds

// ---------------------------------------------------------------------------
// Fused SSIM kernel: one workgroup = one (image, row-strip).
//   Stage A: per-column sliding vertical 8-row sums of {i, t, i^2, t^2, i*t} -> LDS
//   Stage B: horizontal 8-col sums via V_WMMA_F32_16X16X4_F32 against a banded
//            ones matrix, then SSIM formula + masked accumulation.
// ---------------------------------------------------------------------------
extern "C" __global__ __launch_bounds__(512, 1)
void ssim_main(const float* __restrict__ gin, const float* __restrict__ gtg,
               float* __restrict__ partial) {
  extern __shared__ float smem[];
  const int tid  = threadIdx.x;       // 0..511
  const int lane = tid & 31;
  const int wave = tid >> 5;          // 0..15
  const int strip = blockIdx.x & 3;   // 4 strips per image
  const int img   = blockIdx.x >> 2;  // 0..95  (b*3 + c)
  const size_t base = (size_t)img * (W * H);
  const float* gi = gin + base;
  const float* gt = gtg + base;

  const int vr0    = strip * 128;           // first output row of this strip
  const int vr_end = min(vr0 + 128, OH);    // exclusive

  // Constant banded B matrices (K=4 chunks, j=0..5 covering kappa=0..23).
  // B[kappa, n] = 1 iff n <= kappa <= n+7.  K split across half-waves: V0 holds
  // K={0|2}, V1 holds K={1|3} (mirrors documented 32-bit A 16x4 layout).
  v2f Bc[6];
  {
    const int n = lane & 15;
    const int kbase = (lane >> 4) ? 2 : 0;
#pragma unroll
    for (int j = 0; j < 6; ++j) {
      const int K0 = 4 * j + kbase;
      const int K1 = K0 + 1;
      Bc[j].x = (K0 >= n && K0 <= n + 7) ? 1.0f : 0.0f;
      Bc[j].y = (K1 >= n && K1 <= n + 7) ? 1.0f : 0.0f;
    }
  }

  // ---- Stage A state: thread owns column c; 8-deep ring of raw (i,t) ----
  const int c = tid;
  float ring_i[8], ring_t[8];
  float s_i = 0.f, s_t = 0.f, s_ii = 0.f, s_tt = 0.f, s_it = 0.f;
#pragma unroll
  for (int p = 0; p < 8; ++p) { ring_i[p] = 0.f; ring_t[p] = 0.f; }
  // Prime with rows vr0 .. vr0+6  (vr0 % 8 == 0, so row r lives in slot r&7).
#pragma unroll
  for (int p = 0; p < 7; ++p) {
    const int ir = vr0 + p;
    const float a = gi[(size_t)ir * W + c];
    const float b = gt[(size_t)ir * W + c];
    s_i += a; s_t += b; s_ii += a * a; s_tt += b * b; s_it += a * b;
    ring_i[p] = a; ring_t[p] = b;
  }

  float lsum = 0.0f;

  for (int chunk = 0; chunk < 8; ++chunk) {
    const int vrc = vr0 + chunk * 16;

    // ---- Stage A: vertical 8-row box sums for V rows vrc..vrc+15 -> LDS ----
#pragma unroll
    for (int i2 = 0; i2 < 16; ++i2) {
      const int vr = vrc + i2;
      const int ir = min(vr + 7, H - 1);       // clamp; invalid rows masked later
      const float a = gi[(size_t)ir * W + c];
      const float b = gt[(size_t)ir * W + c];
      const float vi  = s_i  + a;
      const float vt  = s_t  + b;
      const float vii = s_ii + a * a;
      const float vtt = s_tt + b * b;
      const float vit = s_it + a * b;
      float* row = smem + i2 * ROWSTRIDE + c;
      row[0 * QSTRIDE] = vi;
      row[1 * QSTRIDE] = vt;
      row[2 * QSTRIDE] = vii;
      row[3 * QSTRIDE] = vtt;
      row[4 * QSTRIDE] = vit;
      // retire oldest row (vr) from the running 7-row sums
      const int so = vr & 7;
      const float oa = ring_i[so], ob = ring_t[so];
      s_i  = vi  - oa;
      s_t  = vt  - ob;
      s_ii = vii - oa * oa;
      s_tt = vtt - ob * ob;
      s_it = vit - oa * ob;
      const int sn = (vr + 7) & 7;
      ring_i[sn] = a; ring_t[sn] = b;
    }
    __syncthreads();

    // ---- Stage B: each wave does two 16-col output tiles via WMMA ----
#pragma unroll
    for (int tt = 0; tt < 2; ++tt) {
      const int tile = wave + tt * 16;   // 0..31
      const int c0 = tile * 16;
      v8f acc[5];
      const v8f vzero = {};
#pragma unroll
      for (int q = 0; q < 5; ++q) acc[q] = vzero;

      const int m  = lane & 15;          // A-matrix row on lanes 0..15
      const int kh = (lane >> 4) * 2;    // K half-wave split
#pragma unroll
      for (int j = 0; j < 6; ++j) {
        const int colA = c0 + 4 * j + kh;
#pragma unroll
        for (int q = 0; q < 5; ++q) {
          const v2f A = *(const v2f*)(smem + q * QSTRIDE + m * ROWSTRIDE + colA);
          acc[q] = __builtin_amdgcn_wmma_f32_16x16x4_f32(
              false, A, false, Bc[j], (short)0, acc[q], false, false);
        }
      }

      // D layout: lane holds (row = vrc + v + 8*(lane>>4), col = c0 + lane&15)
      const int col   = c0 + (lane & 15);
      const int rbase = vrc + ((lane >> 4) ? 8 : 0);
#pragma unroll
      for (int v = 0; v < 8; ++v) {
        const int vr = rbase + v;
        const float mi  = acc[0][v] * INV64;
        const float mt  = acc[1][v] * INV64;
        const float eii = acc[2][v] * INV64;
        const float ett = acc[3][v] * INV64;
        const float eit = acc[4][v] * INV64;
        const float mimt = mi * mt;
        const float mi2 = mi * mi, mt2 = mt * mt;
        const float anum = (2.0f * mimt + SSIM_C1) * (2.0f * (eit - mimt) + SSIM_C2);
        const float bden = (mi2 + mt2 + SSIM_C1) * ((eii - mi2) + (ett - mt2) + SSIM_C2);
        const float val = anum / bden;
        if (col < OW && vr < vr_end) lsum += val;   // branch, so NaN halo never enters
      }
    }
    __syncthreads();
  }

  // ---- block reduction -> one deterministic partial per workgroup ----
  smem[tid] = lsum;
  __syncthreads();
  for (int off = 256; off > 0; off >>= 1) {
    if (tid < off) smem[tid] += smem[tid + off];
    __syncthreads();
  }
  if (tid == 0) partial[blockIdx.x] = smem[0];
}

// partial[img*4 + strip], img = b*3 + c  ->  batch b owns [b*12, b*12+12)
extern "C" __global__ void ssim_finalize(const float* __restrict__ partial,
                                         float* __restrict__ out) {
  const int b = threadIdx.x;
  if (b < 32) {
    float s = 0.0f;
#pragma unroll
    for (int k = 0; k < 12; ++k) s += partial[b * 12 + k];
    out[b] = s * (1.0f / (3.0f * 505.0f * 505.0f));
  }
}

extern "C" void kernel_launch(void* const* d_in, const int* in_sizes, int n_in,
                              void* d_out, int out_size, void* d_ws, size_t ws_size,
                              hipStream_t stream) {
  (void)in_sizes; (void)n_in; (void)out_size; (void)ws_size;
  const float* in = (const float*)d_in[0];
  const float* tg = (const float*)d_in[1];
  float* partial = (float*)d_ws;            // needs 384 floats
  const size_t shmem = (size_t)LDS_FLOATS * sizeof(float);  // ~165 KB of 320 KB WGP LDS

  ssim_main<<<dim3(96 * 4), dim3(512), shmem, stream>>>(in, tg, partial);
  ssim_finalize<<<dim3(1), dim3(32), 0, stream>>>(partial, (float*)d_out);
}